// NeuralODEDecomp_12137577578985
// MI455X (gfx1250) — compile-verified
//
#include <hip/hip_runtime.h>
#include <hip/hip_bf16.h>

// -------- problem constants (match reference) --------
constexpr int B_   = 1024;
constexpr int T_   = 96;
constexpr int NV   = 7;
constexpr int H_   = 64;
constexpr int C_   = 3;
constexpr int H4_  = 256;

// -------- tiling --------
constexpr int BDIM = 256;        // 8 waves (wave32)
constexpr int BT   = 16;         // batch elements per workgroup
constexpr int ROWS = BT * C_;    // 48 state rows (B*C tile)
constexpr int SSTR = 65;         // state LDS stride (bank-conflict free)
constexpr int INK  = 68;         // GEMM1 K padded: 64 + ie + t + 2 pad
constexpr int AST  = 260;        // activation LDS stride
constexpr int HST  = 68;         // decode hidden stride

typedef __attribute__((ext_vector_type(2))) float v2f;
typedef __attribute__((ext_vector_type(8))) float v8f;

__device__ __forceinline__ v8f wmma_f32(v2f a, v2f b, v8f c) {
  // D = A(16x4) * B(4x16) + C(16x16), fp32 matrix pipeline
  return __builtin_amdgcn_wmma_f32_16x16x4_f32(false, a, false, b, (short)0, c,
                                               false, false);
}

__global__ __launch_bounds__(BDIM, 1) void node_persistent_kernel(
    const float* __restrict__ x,
    const float* __restrict__ iw1, const float* __restrict__ ib1,
    const float* __restrict__ iw2, const float* __restrict__ ib2,
    const float* __restrict__ ow1, const float* __restrict__ ob1,
    const float* __restrict__ ow2, const float* __restrict__ ob2,
    const float* __restrict__ inter, const float* __restrict__ cons,
    const float* __restrict__ dw1, const float* __restrict__ db1,
    const float* __restrict__ dw2, const float* __restrict__ db2,
    const float* __restrict__ tp,
    float* __restrict__ out)
{
  __shared__ float sY  [ROWS * SSTR];   // current state y       (48 x 64)
  __shared__ float sYt [ROWS * SSTR];   // RK4 trial state
  __shared__ float sAcc[ROWS * SSTR];   // k1 + 2k2 + 2k3 + k4 accumulator
  __shared__ float sIn [ROWS * INK];    // GEMM1 A: [state | ie | t | 0 0]
  __shared__ float sAct[ROWS * AST];    // tanh activations (48 x 256)
  __shared__ float sHid[C_ * 16 * HST]; // decode hidden (3 x 16 x 64)
  __shared__ float sDW1[C_ * H_ * H_];  // decode W1 (3 x 64 x 64)
  __shared__ float sDW2[C_ * H_ * 16];  // decode W2 padded (3 x 64 x 16)
  __shared__ float sPart[ROWS * 4];     // mean partials
  __shared__ float sSC [ROWS];          // per-row mean over H
  __shared__ float sIE [ROWS];          // interaction effect per row
  __shared__ float sXm [BT * 8];        // x mean over T
  __shared__ float sH0 [BT * H_];       // init hidden
  __shared__ float sOB1[H4_];
  __shared__ float sOB2[H_];
  __shared__ float sDB1[C_ * H_];
  __shared__ float sDB2[C_ * 8];
  __shared__ float sI33[9], sC33[9];

  const int tid  = threadIdx.x;
  const int lane = tid & 31;
  const int wave = tid >> 5;        // 0..7, uniform per wave
  const int hl   = lane >> 4;       // half-wave select (K offset 0 / 2)
  const int ln   = lane & 15;
  const int b0   = blockIdx.x * BT;
  const int n2   = (wave & 3) * 16 + ln;   // this wave's GEMM2 output column

  // ---------------- preload constants into LDS ----------------
  for (int i = tid; i < H4_; i += BDIM) sOB1[i] = ob1[i];
  if (tid < H_) sOB2[tid] = ob2[tid];
  for (int i = tid; i < C_ * H_; i += BDIM) sDB1[i] = db1[i];
  if (tid < C_ * NV) sDB2[(tid / NV) * 8 + (tid % NV)] = db2[tid];
  if (tid < 9) { sI33[tid] = inter[tid]; sC33[tid] = cons[tid]; }
  for (int i = tid; i < C_ * H_ * H_; i += BDIM) sDW1[i] = dw1[i];
  for (int i = tid; i < C_ * H_ * 16; i += BDIM) {
    const int c = i / (H_ * 16), k = (i / 16) % H_, n = i % 16;
    sDW2[i] = (n < NV) ? dw2[((size_t)c * H_ + k) * NV + n] : 0.f;
  }

  // ------- register-resident ode_w1 B-fragments (reused 380x) -------
  // wave owns N-tiles {2w, 2w+1}; reference uses ode_w1[:66] -> rows >=66 are 0
  v2f w1f[2][17];
  #pragma unroll
  for (int j = 0; j < 2; ++j) {
    const int n = (2 * wave + j) * 16 + ln;
    #pragma unroll
    for (int ks = 0; ks < 17; ++ks) {
      const int k = ks * 4 + 2 * hl;
      w1f[j][ks].x = (k     < 66) ? ow1[(size_t)k       * H4_ + n] : 0.f;
      w1f[j][ks].y = (k + 1 < 66) ? ow1[(size_t)(k + 1) * H4_ + n] : 0.f;
    }
  }
  // ------- register-resident ode_w2 B-fragments (column n2, all K) -------
  v2f w2f[64];
  #pragma unroll
  for (int ks = 0; ks < 64; ++ks) {
    const int k = ks * 4 + 2 * hl;
    w2f[ks].x = ow2[(size_t)k       * H_ + n2];
    w2f[ks].y = ow2[(size_t)(k + 1) * H_ + n2];
  }

  // ---------------- init: x mean over T ----------------
  for (int i = tid; i < BT * NV; i += BDIM) {
    const int b = i / NV, v = i % NV;
    const float* xp = x + ((size_t)(b0 + b) * T_) * NV + v;
    float s = 0.f;
    for (int t = 0; t < T_; ++t) s += xp[t * NV];
    sXm[b * 8 + v] = s * (1.f / T_);
  }
  __syncthreads();

  // h0 = relu(xm @ iw1 + ib1)
  for (int i = tid; i < BT * H_; i += BDIM) {
    const int b = i / H_, u = i % H_;
    float s = ib1[u];
    #pragma unroll
    for (int v = 0; v < NV; ++v) s += sXm[b * 8 + v] * iw1[v * H_ + u];
    sH0[i] = fmaxf(s, 0.f);
  }
  __syncthreads();

  // y0 = h0 @ iw2 + ib2  ->  sY rows (b*C + c)
  for (int i = tid; i < BT * C_ * H_; i += BDIM) {
    const int b = i / (C_ * H_), j = i % (C_ * H_);
    float s = ib2[j];
    for (int u = 0; u < H_; ++u) s += sH0[b * H_ + u] * iw2[u * (C_ * H_) + j];
    const int c = j / H_, col = j % H_;
    sY[(b * C_ + c) * SSTR + col] = s;
  }
  __syncthreads();

  // ---------------- decode: per-component MLP via WMMA ----------------
  auto decode = [&](int ti) {
    // GEMM1: hid = relu(Y_c(16x64) @ dw1[c](64x64) + db1[c]); 12 jobs / 8 waves
    for (int j = wave; j < 12; j += 8) {           // wave-uniform
      const int c = j >> 2, nt = j & 3;
      v8f acc = {};
      const int n = nt * 16 + ln;
      #pragma unroll
      for (int ks = 0; ks < 16; ++ks) {
        const int k = ks * 4 + 2 * hl;
        v2f a, bf;
        const int r = (ln * C_ + c) * SSTR;
        a.x  = sY[r + k];                  a.y  = sY[r + k + 1];
        bf.x = sDW1[c * 4096 + k * H_ + n]; bf.y = sDW1[c * 4096 + (k + 1) * H_ + n];
        acc = wmma_f32(a, bf, acc);
      }
      const float bb = sDB1[c * H_ + n];
      #pragma unroll
      for (int r = 0; r < 8; ++r) {
        const int m = r + 8 * hl;
        sHid[(c * 16 + m) * HST + n] = fmaxf(acc[r] + bb, 0.f);
      }
    }
    __syncthreads();
    // GEMM2: (16x64) @ dw2[c](64x7 padded to 16) -> out[(c, b, ti, :7)]
    if (wave < C_) {                                // wave-uniform
      const int c = wave;
      v8f acc = {};
      #pragma unroll
      for (int ks = 0; ks < 16; ++ks) {
        const int k = ks * 4 + 2 * hl;
        v2f a, bf;
        a.x  = sHid[(c * 16 + ln) * HST + k];
        a.y  = sHid[(c * 16 + ln) * HST + k + 1];
        bf.x = sDW2[c * 1024 + k * 16 + ln];
        bf.y = sDW2[c * 1024 + (k + 1) * 16 + ln];
        acc = wmma_f32(a, bf, acc);
      }
      if (ln < NV) {
        const float bb = sDB2[c * 8 + ln];
        #pragma unroll
        for (int r = 0; r < 8; ++r) {
          const int m = r + 8 * hl;
          out[(((size_t)c * B_ + (b0 + m)) * T_ + ti) * NV + ln] = acc[r] + bb;
        }
      }
    }
    __syncthreads();
  };

  // ---------------- ODE function f(t, S) fused with RK4 combine ----------------
  // mode 0: k1 -> sAcc = d;        sYt = y + 0.5h d
  // mode 1: k2 -> sAcc += 2d;      sYt = y + 0.5h d
  // mode 2: k3 -> sAcc += 2d;      sYt = y +  h  d
  // mode 3: k4 -> y += (h/6)(sAcc + d)
  auto odeF = [&](float tval, const float* S, int mode, float hstep) {
    // per-row mean over H (4-way parallel reduction)
    for (int i = tid; i < ROWS * 4; i += BDIM) {
      const int r = i >> 2, q = i & 3;
      float s = 0.f;
      #pragma unroll
      for (int k = 0; k < 16; ++k) s += S[r * SSTR + q * 16 + k];
      sPart[i] = s;
    }
    __syncthreads();
    for (int r = tid; r < ROWS; r += BDIM)
      sSC[r] = (sPart[4 * r] + sPart[4 * r + 1] + sPart[4 * r + 2] + sPart[4 * r + 3]) * (1.f / H_);
    __syncthreads();
    // ie = conservation @ (interaction @ mean)
    if (tid < BT) {
      const float m0 = sSC[tid * 3 + 0], m1 = sSC[tid * 3 + 1], m2 = sSC[tid * 3 + 2];
      const float e0 = sI33[0] * m0 + sI33[1] * m1 + sI33[2] * m2;
      const float e1 = sI33[3] * m0 + sI33[4] * m1 + sI33[5] * m2;
      const float e2 = sI33[6] * m0 + sI33[7] * m1 + sI33[8] * m2;
      sIE[tid * 3 + 0] = sC33[0] * e0 + sC33[1] * e1 + sC33[2] * e2;
      sIE[tid * 3 + 1] = sC33[3] * e0 + sC33[4] * e1 + sC33[5] * e2;
      sIE[tid * 3 + 2] = sC33[6] * e0 + sC33[7] * e1 + sC33[8] * e2;
    }
    __syncthreads();
    // build GEMM1 A: [state(64) | ie | t | 0 | 0]
    for (int i = tid; i < ROWS * INK; i += BDIM) {
      const int r = i / INK, k = i % INK;
      float vv;
      if      (k <  H_)     vv = S[r * SSTR + k];
      else if (k == H_)     vv = sIE[r];
      else if (k == H_ + 1) vv = tval;
      else                  vv = 0.f;
      sIn[i] = vv;
    }
    __syncthreads();

    // GEMM1: (48 x 68) @ (68 x 256); W1 frags register-resident, no VMEM
    v8f acc1[3][2] = {};
    #pragma unroll
    for (int ks = 0; ks < 17; ++ks) {
      const int k = ks * 4 + 2 * hl;
      v2f a[3];
      #pragma unroll
      for (int mt = 0; mt < 3; ++mt) {
        const int r = (mt * 16 + ln) * INK;
        a[mt].x = sIn[r + k];  a[mt].y = sIn[r + k + 1];
      }
      #pragma unroll
      for (int j = 0; j < 2; ++j) {
        acc1[0][j] = wmma_f32(a[0], w1f[j][ks], acc1[0][j]);
        acc1[1][j] = wmma_f32(a[1], w1f[j][ks], acc1[1][j]);
        acc1[2][j] = wmma_f32(a[2], w1f[j][ks], acc1[2][j]);
      }
    }
    // bias + tanh -> sAct
    #pragma unroll
    for (int j = 0; j < 2; ++j) {
      const int n = (2 * wave + j) * 16 + ln;
      const float bb = sOB1[n];
      #pragma unroll
      for (int mt = 0; mt < 3; ++mt)
        #pragma unroll
        for (int r = 0; r < 8; ++r) {
          const int m = mt * 16 + r + 8 * hl;
          sAct[m * AST + n] = tanhf(acc1[mt][j][r] + bb);
        }
    }
    __syncthreads();

    // GEMM2: (48 x 256) @ ode_w2(256 x 64); W2 frags register-resident, no VMEM
    auto gemm2_pair = [&](int mt) {
      v8f acc = {};
      const int arow = (mt * 16 + ln) * AST;
      #pragma unroll
      for (int ks = 0; ks < 64; ++ks) {
        const int k = ks * 4 + 2 * hl;
        v2f a;
        a.x = sAct[arow + k];  a.y = sAct[arow + k + 1];
        acc = wmma_f32(a, w2f[ks], acc);
      }
      const float bb = sOB2[n2];
      #pragma unroll
      for (int r = 0; r < 8; ++r) {
        const int m = mt * 16 + r + 8 * hl;
        const float d = acc[r] + bb;
        const int idx = m * SSTR + n2;
        if (mode == 0)      { sAcc[idx] = d;          sYt[idx] = sY[idx] + 0.5f * hstep * d; }
        else if (mode == 1) { sAcc[idx] += 2.f * d;   sYt[idx] = sY[idx] + 0.5f * hstep * d; }
        else if (mode == 2) { sAcc[idx] += 2.f * d;   sYt[idx] = sY[idx] + hstep * d; }
        else                { sY[idx] += (hstep * (1.f / 6.f)) * (sAcc[idx] + d); }
      }
    };
    gemm2_pair(wave >> 2);        // mt = 0 or 1  (all waves)
    if (wave < 4) gemm2_pair(2);  // mt = 2       (waves 0..3), wave-uniform
    __syncthreads();
  };

  // ---------------- time integration ----------------
  decode(0);
  #pragma unroll 1
  for (int ti = 1; ti < T_; ++ti) {
    const float t0 = tp[ti - 1];
    const float t1 = tp[ti];
    const float h  = t1 - t0;
    #pragma unroll 1
    for (int sub = 0; sub < 4; ++sub) {   // single inlined odeF copy
      const float tv = (sub == 0) ? t0 : ((sub == 3) ? t1 : t0 + 0.5f * h);
      const float* S = (sub == 0) ? sY : sYt;
      odeF(tv, S, sub, h);
    }
    decode(ti);
  }
}

extern "C" void kernel_launch(void* const* d_in, const int* in_sizes, int n_in,
                              void* d_out, int out_size, void* d_ws, size_t ws_size,
                              hipStream_t stream) {
  const float* x     = (const float*)d_in[0];
  const float* iw1   = (const float*)d_in[1];
  const float* ib1   = (const float*)d_in[2];
  const float* iw2   = (const float*)d_in[3];
  const float* ib2   = (const float*)d_in[4];
  const float* ow1   = (const float*)d_in[5];
  const float* ob1   = (const float*)d_in[6];
  const float* ow2   = (const float*)d_in[7];
  const float* ob2   = (const float*)d_in[8];
  const float* inter = (const float*)d_in[9];
  const float* cons  = (const float*)d_in[10];
  const float* dw1   = (const float*)d_in[11];
  const float* db1   = (const float*)d_in[12];
  const float* dw2   = (const float*)d_in[13];
  const float* db2   = (const float*)d_in[14];
  const float* tp    = (const float*)d_in[15];
  float* out = (float*)d_out;

  node_persistent_kernel<<<dim3(B_ / BT), dim3(BDIM), 0, stream>>>(
      x, iw1, ib1, iw2, ib2, ow1, ob1, ow2, ob2, inter, cons,
      dw1, db1, dw2, db2, tp, out);
}